// MarchingCubeHelper_7275674600181
// MI455X (gfx1250) — compile-verified
//
#include <hip/hip_runtime.h>
#include <hip/hip_bf16.h>
#include <stdint.h>

// Marching tetrahedra, RES=64 level set.
// Store-bandwidth-bound: ~156 MB out vs 1 MB in -> lower bound ~6.7us @ 23.3 TB/s.
// CDNA5 path: async global->LDS staging of the per-block 2x2x64 slab
// (global_load_async_to_lds_b128 + s_wait_asynccnt), corner gathers from LDS,
// wide b64 stores for the output stream.

#define RES   64
#define RC    (RES - 1)          // 63 cells per axis
#define NCELL (RC * RC * RC)     // 250047
#define NTET  (NCELL * 6)        // 1500282

__device__ __constant__ int c_tets[6][4] = {
    {0,5,1,6},{0,1,2,6},{0,2,3,6},{0,3,7,6},{0,7,4,6},{0,4,5,6}};
__device__ __constant__ int c_cube[8][3] = {
    {0,0,0},{1,0,0},{1,1,0},{0,1,0},{0,0,1},{1,0,1},{1,1,1},{0,1,1}};
__device__ __constant__ int c_edges[6][2] = {
    {0,1},{1,2},{2,0},{0,3},{1,3},{2,3}};
// TET_TRIS flattened to 16 x 6 (two triangles x 3 edge ids, -1 = invalid)
__device__ __constant__ int c_tris[16][6] = {
    {-1,-1,-1,-1,-1,-1},
    { 0, 3, 2,-1,-1,-1},
    { 0, 1, 4,-1,-1,-1},
    { 2, 3, 4, 2, 4, 1},
    { 1, 5, 2,-1,-1,-1},
    { 0, 1, 5, 0, 5, 3},
    { 0, 4, 5, 0, 5, 2},
    { 3, 4, 5,-1,-1,-1},
    { 3, 5, 4,-1,-1,-1},
    { 0, 2, 5, 0, 5, 4},
    { 0, 3, 5, 0, 5, 1},
    { 1, 2, 5,-1,-1,-1},
    { 1, 4, 3, 1, 3, 2},
    { 0, 4, 1,-1,-1,-1},
    { 0, 2, 3,-1,-1,-1},
    {-1,-1,-1,-1,-1,-1}};

__global__ __launch_bounds__(384)
void mt_kernel(const float* __restrict__ level, float* __restrict__ out)
{
    // LDS tile: level[i:i+2][j:j+2][0:64] -> 2*2*64 = 256 f32 = 1 KB
    __shared__ float tile[256];

    const int bid = blockIdx.x;
    const int i   = bid / RC;        // cell i (0..62)
    const int j   = bid % RC;        // cell j (0..62)
    const int tid = threadIdx.x;

    // ---- async stage of the slab into LDS: 64 lanes x b128 (16 B each) ----
    // Slab rows are 256B-aligned in global and LDS, so every 16B chunk is
    // 16B-aligned on both sides.
    if (tid < 64) {
        const int di  = tid >> 5;          // 0..1
        const int dj  = (tid >> 4) & 1;    // 0..1
        const int kk4 = (tid & 15) * 4;    // 0,4,...,60  (4 floats per lane)
        const uint32_t goff = (uint32_t)((((i + di) * RES + (j + dj)) * RES + kk4) * 4);
        // low 32 bits of the flat shared address == wave-relative LDS byte offset
        const uint32_t loff = (uint32_t)(uintptr_t)&tile[tid * 4];
        asm volatile("global_load_async_to_lds_b128 %0, %1, %2"
                     :: "v"(loff), "v"(goff),
                        "s"((uint64_t)(uintptr_t)level)
                     : "memory");
    }
    asm volatile("s_wait_asynccnt 0" ::: "memory");
    __syncthreads();

    if (tid >= RC * 6) return;       // 378 active tet-threads per block

    const int kc = tid / 6;          // cell k (0..62)
    const int t  = tid - kc * 6;     // tet id (0..5)

    // cube corner values, f = -level
    float fc[8];
#pragma unroll
    for (int n = 0; n < 8; ++n) {
        const int lidx = ((c_cube[n][0] * 2 + c_cube[n][1]) << 6) + c_cube[n][2] + kc;
        fc[n] = -tile[lidx];
    }

    // tet vertex values and integer lattice positions (i,j,k order)
    float fv[4];
    float px[4], py[4], pz[4];
#pragma unroll
    for (int vtx = 0; vtx < 4; ++vtx) {
        const int cid = c_tets[t][vtx];
        fv[vtx] = fc[cid];
        px[vtx] = (float)(i  + c_cube[cid][0]);
        py[vtx] = (float)(j  + c_cube[cid][1]);
        pz[vtx] = (float)(kc + c_cube[cid][2]);
    }

    const int cse = (fv[0] > 0.f ? 1 : 0) | (fv[1] > 0.f ? 2 : 0)
                  | (fv[2] > 0.f ? 4 : 0) | (fv[3] > 0.f ? 8 : 0);

    const int c = (i * RC + j) * RC + kc;   // cell index (matches reshape order)
    const int g = c * 6 + t;                // global tet index

    // ---- 6 edge vertices -> 18 f32, coords reversed (k,j,i), /(RES-1) ----
    const float inv = 1.0f / (float)(RES - 1);
    float vout[18];
#pragma unroll
    for (int e = 0; e < 6; ++e) {
        const int a = c_edges[e][0], b = c_edges[e][1];
        const float fa = fv[a], fb = fv[b];
        float d = fa - fb;
        d = (fabsf(d) < 1e-8f) ? 1e-8f : d;
        float tt = fa / d;
        tt = fminf(fmaxf(tt, 0.f), 1.f);
        const float vx = px[a] + tt * (px[b] - px[a]);  // i coord
        const float vy = py[a] + tt * (py[b] - py[a]);  // j coord
        const float vz = pz[a] + tt * (pz[b] - pz[a]);  // k coord
        vout[e * 3 + 0] = vz * inv;   // reversed last axis: (k, j, i)
        vout[e * 3 + 1] = vy * inv;
        vout[e * 3 + 2] = vx * inv;
    }
    // 72 B per tet, 8B-aligned -> 9x global_store_b64
    float2* vdst = (float2*)(out + (size_t)g * 18);
#pragma unroll
    for (int q = 0; q < 9; ++q)
        vdst[q] = make_float2(vout[2 * q], vout[2 * q + 1]);

    // ---- triangles: 6 i32 per tet, region starts after NTET*18 f32 ----
    const int basee = g * 6;
    int2* tdst = (int2*)((int*)(out + (size_t)NTET * 18) + (size_t)g * 6);
#pragma unroll
    for (int q = 0; q < 3; ++q) {
        const int e0 = c_tris[cse][2 * q + 0];
        const int e1 = c_tris[cse][2 * q + 1];
        tdst[q] = make_int2(basee + max(e0, 0), basee + max(e1, 0));
    }

    // ---- valid flags: 2 per tet, stored as i32 0/1 after the tri region ----
    int2* fdst = (int2*)((int*)(out + (size_t)NTET * 18 + (size_t)NTET * 6)
                         + (size_t)g * 2);
    fdst[0] = make_int2(c_tris[cse][0] >= 0 ? 1 : 0,
                        c_tris[cse][3] >= 0 ? 1 : 0);
}

extern "C" void kernel_launch(void* const* d_in, const int* in_sizes, int n_in,
                              void* d_out, int out_size, void* d_ws, size_t ws_size,
                              hipStream_t stream) {
    (void)in_sizes; (void)n_in; (void)out_size; (void)d_ws; (void)ws_size;
    const float* level = (const float*)d_in[0];
    float* out = (float*)d_out;
    // one block per (i,j) cell row: 63*63 blocks x 384 threads (12 wave32s)
    mt_kernel<<<dim3(RC * RC), dim3(384), 0, stream>>>(level, out);
}